// GaussianLayer_11673721110546
// MI455X (gfx1250) — compile-verified
//
#include <hip/hip_runtime.h>
#include <cstdint>
#include <cstddef>

#define B_    4
#define L_    1024
#define KNN_  32
#define A_    5
#define K_    16
#define MAXAA_ 22
#define NROWS (KNN_ * A_)        /* 160 neighbor-atom rows per (b,l) */
#define NPAIR (A_ * A_)          /* 25 */
#define GBF_PER (KNN_ * NPAIR * K_)  /* 12800 */
#define GBF_TOTAL ((size_t)B_ * L_ * GBF_PER)

#define USE_ASYNC_LDS_ASM 1

typedef __attribute__((ext_vector_type(2))) float v2f;
typedef __attribute__((ext_vector_type(8))) float v8f;

__global__ __launch_bounds__(128)
void gaussian_layer_kernel(const int*   __restrict__ aa,
                           const float* __restrict__ X,
                           const int*   __restrict__ E_idx,
                           const float* __restrict__ mask_atoms,
                           const float* __restrict__ mask_attend,
                           const float* __restrict__ means,
                           const float* __restrict__ stds,
                           const float* __restrict__ mul_w,
                           const float* __restrict__ bias_w,
                           const float* __restrict__ aa_embed,
                           float*       __restrict__ out)
{
    __shared__ float p_aos[NROWS][4];     // x,y,z,|p|^2 per neighbor-atom row
    __shared__ float msm[NROWS];          // neighbor atom mask per row
    __shared__ int   kn5sm[NROWS];        // (row/5)*5 lookup (kills divisions later)
    __shared__ float qv[3][16];           // center atom coords (padded to 16)
    __shared__ float qq[16];              // |q|^2 (padded to 16)
    __shared__ float mul_sm[NPAIR], bias_sm[NPAIR];
    __shared__ float matt[KNN_];
    __shared__ int   aan[KNN_];
    __shared__ int   aac_sm;
    __shared__ float meank[K_], hk_sm[K_], coefk[K_];   // hk = -0.5/std^2
    __shared__ float Dsm[KNN_ * NPAIR];   // masked affine(D), index rv*5 + a2

    const int tid = threadIdx.x;
    const int bl  = blockIdx.x;           // b*L + l
    const int b   = bl >> 10;             // L = 1024

    // ------------------ Phase 0: stage all per-(b,l) data into LDS ------------------
    for (int r = tid; r < NROWS; r += 128) {
        int kn = r / A_, a1 = r - kn * A_;
        kn5sm[r] = kn * A_;
        int e  = E_idx[(size_t)bl * KNN_ + kn];
        size_t xoff = (((size_t)b * L_ + e) * A_ + a1) * 3;
        msm[r] = mask_atoms[((size_t)b * L_ + e) * A_ + a1];
#if USE_ASYNC_LDS_ASM
        {   // CDNA5 async DMA global->LDS (ASYNCcnt); inst offset applies to both
            // the LDS and global addresses, so one pair serves x/y/z.
            uint32_t lds_b = (uint32_t)(uintptr_t)&p_aos[r][0];
            const float* gsrc = X + xoff;
            asm volatile("global_load_async_to_lds_b32 %0, %1, off"
                         :: "v"(lds_b), "v"(gsrc) : "memory");
            asm volatile("global_load_async_to_lds_b32 %0, %1, off offset:4"
                         :: "v"(lds_b), "v"(gsrc) : "memory");
            asm volatile("global_load_async_to_lds_b32 %0, %1, off offset:8"
                         :: "v"(lds_b), "v"(gsrc) : "memory");
        }
#else
        float x = X[xoff + 0], y = X[xoff + 1], z = X[xoff + 2];
        p_aos[r][0] = x; p_aos[r][1] = y; p_aos[r][2] = z;
#endif
    }
    if (tid < A_) {
        size_t qoff = ((size_t)bl * A_ + tid) * 3;
        float x = X[qoff], y = X[qoff + 1], z = X[qoff + 2];
        qv[0][tid] = x; qv[1][tid] = y; qv[2][tid] = z;
        qq[tid] = x * x + y * y + z * z;
    }
    if (tid >= A_ && tid < 16) {          // zero pads so WMMA N-columns 5..15 contribute 0
        qv[0][tid] = 0.f; qv[1][tid] = 0.f; qv[2][tid] = 0.f; qq[tid] = 0.f;
    }
    if (tid < K_) {
        float s  = fabsf(stds[tid]) + 0.01f;
        float is = 1.0f / s;
        meank[tid] = means[tid];
        hk_sm[tid] = -0.5f * is * is;
        coefk[tid] = 1.0f / (2.50662827f * s);   // 1/(sqrt(2*3.1415926)*std)
    }
    if (tid < NPAIR) { mul_sm[tid] = mul_w[tid]; bias_sm[tid] = bias_w[tid]; }
    if (tid < KNN_) {
        matt[tid] = mask_attend[(size_t)bl * KNN_ + tid];
        int e = E_idx[(size_t)bl * KNN_ + tid];
        aan[tid] = aa[(size_t)b * L_ + e];
    }
    if (tid == 0) aac_sm = aa[bl];
#if USE_ASYNC_LDS_ASM
    asm volatile("s_wait_asynccnt 0x0" ::: "memory");
#endif
    __syncthreads();

    // Phase 0b: |p|^2 per row (coords now resident in LDS)
    for (int r = tid; r < NROWS; r += 128) {
        float x = p_aos[r][0], y = p_aos[r][1], z = p_aos[r][2];
        p_aos[r][3] = x * x + y * y + z * z;
    }
    __syncthreads();

    // ------------------ Phase 1: distances via V_WMMA_F32_16X16X4_F32 ------------------
    // sq[m][n] = |p_m|^2 + |q_n|^2 - 2 p_m.q_n  ==  WMMA(A=p, B=-2q, C=|p|^2+|q|^2)
    const int wave = tid >> 5;
    const int lane = tid & 31;
    const int n    = lane & 15;   // A-row (M) index within chunk AND B/C column (N)
    const int hi   = lane >> 4;   // selects K={0,1} vs K={2,3} halves

    v2f Bv;                        // B matrix 4x16: column n, scaled by -2
    Bv.x = hi ? (-2.0f * qv[2][n]) : (-2.0f * qv[0][n]);
    Bv.y = hi ? 0.0f               : (-2.0f * qv[1][n]);
    const float qqn = qq[n];

    for (int chunk = wave; chunk < NROWS / 16; chunk += 4) {   // 10 chunks over 4 waves
        int row = chunk * 16 + n;                              // A-matrix row (M = lane&15)
        v2f Av;
        Av.x = hi ? p_aos[row][2] : p_aos[row][0];
        Av.y = hi ? 0.0f          : p_aos[row][1];

        v8f Cv;
#pragma unroll
        for (int v = 0; v < 8; ++v) {
            int rv = chunk * 16 + v + 8 * hi;                  // C/D row M = v + 8*hi
            Cv[v] = p_aos[rv][3] + qqn;
        }

        v8f Dv = __builtin_amdgcn_wmma_f32_16x16x4_f32(
            /*neg_a=*/false, Av, /*neg_b=*/false, Bv,
            /*c_mod=*/(short)0, Cv, /*reuse_a=*/false, /*reuse_b=*/false);

        if (n < A_) {                                          // only 5 real columns
#pragma unroll
            for (int v = 0; v < 8; ++v) {
                int rv  = chunk * 16 + v + 8 * hi;
                int kn5 = kn5sm[rv];                           // division-free indices
                int ci  = (rv - kn5) * A_ + n;                 // a1*5 + a2
                float sq = Dv[v];
                float D  = sq > 0.0f ? __builtin_amdgcn_sqrtf(sq) : 0.0f; // raw v_sqrt_f32
                float x  = mul_sm[ci] * D + bias_sm[ci];
                x *= msm[rv] * msm[kn5 + n];                   // m_nb[a1]*m_nb[a2]
                Dsm[rv * A_ + n] = x;                          // == kn*25 + a1*5 + a2
            }
        }
    }
    __syncthreads();

    // ------------------ Phase 2: Gaussian basis, float4 (b128) stores ------------------
    // 4 threads cover one pair's 16 k-channels; a wave stores 512 contiguous bytes/iter.
    const int kq  = (tid & 3) * 4;      // k base: 0,4,8,12
    const int pgr = tid >> 2;           // 32 pair-groups
    float mk[4], hk[4], ck[4];
#pragma unroll
    for (int j = 0; j < 4; ++j) {
        mk[j] = meank[kq + j]; hk[j] = hk_sm[kq + j]; ck[j] = coefk[kq + j];
    }
    const size_t base_g = (size_t)bl * GBF_PER;
    for (int pair = pgr; pair < KNN_ * NPAIR; pair += 32) {    // 25 iters, uniform
        int   kn = pair / NPAIR;
        float x  = Dsm[pair];
        float ma = matt[kn];
        float4 o;
        { float d = x - mk[0]; o.x = __expf(hk[0] * d * d) * ck[0] * ma; }
        { float d = x - mk[1]; o.y = __expf(hk[1] * d * d) * ck[1] * ma; }
        { float d = x - mk[2]; o.z = __expf(hk[2] * d * d) * ck[2] * ma; }
        { float d = x - mk[3]; o.w = __expf(hk[3] * d * d) * ck[3] * ma; }
        *reinterpret_cast<float4*>(out + base_g + (size_t)pair * K_ + kq) = o;
    }

    // ------------------ Phase 3: aa-pair embedding output (one b128 per thread) ------------------
    const int aac = aac_sm;
    const size_t base_f = GBF_TOTAL + (size_t)bl * (KNN_ * K_);
    {
        int kn = tid >> 2;              // 128 threads == 512 floats exactly
        int kk = (tid & 3) * 4;
        const float4 v = *reinterpret_cast<const float4*>(
            aa_embed + (size_t)(aac * MAXAA_ + aan[kn]) * K_ + kk);
        float ma = matt[kn];
        float4 o = make_float4(v.x * ma, v.y * ma, v.z * ma, v.w * ma);
        *reinterpret_cast<float4*>(out + base_f + (size_t)tid * 4) = o;
    }
}

extern "C" void kernel_launch(void* const* d_in, const int* in_sizes, int n_in,
                              void* d_out, int out_size, void* d_ws, size_t ws_size,
                              hipStream_t stream) {
    (void)in_sizes; (void)n_in; (void)out_size; (void)d_ws; (void)ws_size;
    const int*   aa     = (const int*)  d_in[0];
    const float* X      = (const float*)d_in[1];
    const int*   E_idx  = (const int*)  d_in[2];
    const float* m_at   = (const float*)d_in[3];
    const float* m_att  = (const float*)d_in[4];
    const float* means  = (const float*)d_in[5];
    const float* stds   = (const float*)d_in[6];
    const float* mul_w  = (const float*)d_in[7];
    const float* bias_w = (const float*)d_in[8];
    const float* aa_emb = (const float*)d_in[9];
    float* out = (float*)d_out;

    dim3 grid(B_ * L_);
    dim3 block(128);
    hipLaunchKernelGGL(gaussian_layer_kernel, grid, block, 0, stream,
                       aa, X, E_idx, m_at, m_att, means, stds, mul_w, bias_w, aa_emb, out);
}